// CostModule_65919158059044
// MI455X (gfx1250) — compile-verified
//
#include <hip/hip_runtime.h>
#include <math.h>

#define BB 16
#define NN 256
#define RR 24
#define LL 32
#define STOPT 60.0f
#define XFERT 300.0f
#define SCALARS 144          // floats before trip_times region in d_out
#define INF_BITS 0x7F800000u

typedef float v2f __attribute__((ext_vector_type(2)));
typedef float v8f __attribute__((ext_vector_type(8)));

// ---------------------------------------------------------------------------
// Full 32-lane all-reduce (sum) built from v_wmma_f32_16x16x4_f32.
// Stage 1: values in A (A[m][0]=x[m], A[m][2]=x[m+16], rest 0), B = ones:
//          D1[m][n] = x[m] + x[m+16]  (replicated across n)
// Stage 2: A = ones, B = successive VGPR pairs of D1, C accumulated:
//          D[m][n] = sum over all 16 rows of D1 = total, in every lane.
// Exact f32 (all products are value*1.0), deterministic.
// ---------------------------------------------------------------------------
__device__ __forceinline__ float wmma_allreduce_f32(float x) {
  v2f a;    a[0] = x;    a[1] = 0.0f;
  v2f ones; ones[0] = 1.0f; ones[1] = 1.0f;
  v8f z = {0.f, 0.f, 0.f, 0.f, 0.f, 0.f, 0.f, 0.f};
  v8f d1 = __builtin_amdgcn_wmma_f32_16x16x4_f32(false, a, false, ones,
                                                 (short)0, z, false, false);
  v8f acc = z;
  v2f p;
  p[0] = d1[0]; p[1] = d1[1];
  acc = __builtin_amdgcn_wmma_f32_16x16x4_f32(false, ones, false, p,
                                              (short)0, acc, false, false);
  p[0] = d1[2]; p[1] = d1[3];
  acc = __builtin_amdgcn_wmma_f32_16x16x4_f32(false, ones, false, p,
                                              (short)0, acc, false, false);
  p[0] = d1[4]; p[1] = d1[5];
  acc = __builtin_amdgcn_wmma_f32_16x16x4_f32(false, ones, false, p,
                                              (short)0, acc, false, false);
  p[0] = d1[6]; p[1] = d1[7];
  acc = __builtin_amdgcn_wmma_f32_16x16x4_f32(false, ones, false, p,
                                              (short)0, acc, false, false);
  return acc[0];
}

// ---------------------------------------------------------------------------
// 1) init: dist region <- +INF bit pattern; scalar region <- 0
// ---------------------------------------------------------------------------
__global__ void init_kernel(float* __restrict__ out,
                            unsigned* __restrict__ dist_bits) {
  int i = blockIdx.x * blockDim.x + threadIdx.x;
  if (i < SCALARS) out[i] = 0.0f;
  if (i < BB * NN * NN) dist_bits[i] = INF_BITS;
}

// ---------------------------------------------------------------------------
// 2) routes: per (b,r) block of 32 lanes. Cumulative leg times, then
//    scatter-min all (s<t) pair times into edge[b,src,dst] / edge[b,dst,src].
//    atomicMin on uint bits == float min for non-negative values.
// ---------------------------------------------------------------------------
__global__ __launch_bounds__(32) void routes_kernel(
    const float* __restrict__ dt, const int* __restrict__ routes,
    unsigned* __restrict__ edge_bits) {
  const int b = blockIdx.x / RR;
  const int r = blockIdx.x % RR;
  const int* nd = routes + (b * RR + r) * LL;

  __shared__ int   node[LL];
  __shared__ float cf[LL];   // cumulative forward
  __shared__ float cr[LL];   // cumulative reverse
  __shared__ float lf[LL];
  __shared__ float lr[LL];

  const int lane = threadIdx.x;
  node[lane] = nd[lane];
  __syncthreads();

  if (lane < LL - 1) {
    int s = node[lane], t = node[lane + 1];
    lf[lane] = dt[((size_t)b * NN + s) * NN + t] + STOPT;
    lr[lane] = dt[((size_t)b * NN + t) * NN + s] + STOPT;
  }
  __syncthreads();

  if (lane == 0) {  // serial prefix: matches jnp.cumsum order exactly
    float af = 0.0f, ar = 0.0f;
    cf[0] = 0.0f; cr[0] = 0.0f;
    for (int s = 0; s < LL - 1; ++s) {
      af += lf[s]; ar += lr[s];
      cf[s + 1] = af; cr[s + 1] = ar;
    }
  }
  __syncthreads();

  const int s = lane;
  const int ns = node[s];
  for (int t = s + 1; t < LL; ++t) {
    const int nt = node[t];
    float tf = cf[t] - cf[s];
    float tr = cr[t] - cr[s];
    atomicMin(&edge_bits[((size_t)b * NN + ns) * NN + nt], __float_as_uint(tf));
    atomicMin(&edge_bits[((size_t)b * NN + nt) * NN + ns], __float_as_uint(tr));
  }
}

// ---------------------------------------------------------------------------
// 3) route_time: deterministic (fixed-order) per-batch sum of all legs
// ---------------------------------------------------------------------------
__global__ __launch_bounds__(32) void route_time_kernel(
    const float* __restrict__ dt, const int* __restrict__ routes,
    float* __restrict__ rt_out) {
  const int b = blockIdx.x;
  __shared__ float per_route[RR];
  const int r = threadIdx.x;
  if (r < RR) {
    const int* nd = routes + (b * RR + r) * LL;
    float acc = 0.0f;
    for (int s = 0; s < LL - 1; ++s) {
      int a = nd[s], c = nd[s + 1];
      acc += dt[((size_t)b * NN + a) * NN + c] + STOPT;   // fwd
      acc += dt[((size_t)b * NN + c) * NN + a] + STOPT;   // rev
    }
    per_route[r] = acc;
  }
  __syncthreads();
  if (threadIdx.x == 0) {
    float tot = 0.0f;
    for (int i = 0; i < RR; ++i) tot += per_route[i];
    rt_out[b] = tot;
  }
}

// ---------------------------------------------------------------------------
// 4) Floyd–Warshall, one workgroup per batch, fully LDS-resident:
//    dist f32 256x256 (256KB) + hops u8 (64KB) = 320KB = CDNA5 WG max.
//    Thread t owns row t/4, 64 contiguous cols. At iter k neither row k
//    nor col k changes (alt = d+0), so in-place update + barrier is safe.
// ---------------------------------------------------------------------------
__global__ __launch_bounds__(1024) void fw_kernel(
    float* __restrict__ dist_g, unsigned char* __restrict__ hops_g) {
  __shared__ float         sd[NN * NN];
  __shared__ unsigned char sh[NN * NN];

  const int b = blockIdx.x;
  float* dg = dist_g + (size_t)b * NN * NN;

  for (int t = threadIdx.x; t < NN * NN; t += 1024) {
    float v = dg[t];
    int i = t >> 8, j = t & (NN - 1);
    if (i == j) v = 0.0f;
    sd[t] = v;
    sh[t] = (unsigned char)((i != j && !isinf(v)) ? 1 : 0);
  }
  __syncthreads();

  const int row = threadIdx.x >> 2;
  const int cb  = (threadIdx.x & 3) << 6;

  for (int k = 0; k < NN; ++k) {
    const float dik = sd[row * NN + k];
    if (!isinf(dik)) {
      const int hik = sh[row * NN + k];
      #pragma unroll 4
      for (int jj = 0; jj < 64; ++jj) {
        const int j = cb + jj;
        const float alt = dik + sd[k * NN + j];
        const int idx = row * NN + j;
        if (alt < sd[idx]) {
          sd[idx] = alt;
          sh[idx] = (unsigned char)(hik + sh[k * NN + j]);
        }
      }
    }
    __syncthreads();
  }

  for (int t = threadIdx.x; t < NN * NN; t += 1024) {
    dg[t] = sd[t];
    hops_g[(size_t)b * NN * NN + t] = sh[t];
  }
}

// ---------------------------------------------------------------------------
// 5) finalize: trip_times in place + 9 demand-weighted reductions per batch.
//    Per-wave reduction via WMMA all-reduce, cross-wave via LDS.
// ---------------------------------------------------------------------------
__global__ __launch_bounds__(256) void finalize_kernel(
    const float* __restrict__ demand, float* __restrict__ out,
    const unsigned char* __restrict__ hops) {
  const int b = blockIdx.x;
  float* dist = out + SCALARS + (size_t)b * NN * NN;
  const float* dem = demand + (size_t)b * NN * NN;
  const unsigned char* hp = hops + (size_t)b * NN * NN;

  float a_dt = 0.f, a_tr = 0.f, a_un = 0.f, a_dem = 0.f, a_nd = 0.f;
  float a_b0 = 0.f, a_b1 = 0.f, a_b2 = 0.f, a_bu = 0.f;

  for (int t = threadIdx.x; t < NN * NN; t += 256) {
    const float d  = dist[t];
    const float de = dem[t];
    const float h  = (float)hp[t];
    const bool nopath = isinf(d);
    const float pl = nopath ? 0.0f : h + 1.0f;
    const float tr = (pl == 0.0f) ? 0.0f : pl - 2.0f;
    const float tt = nopath ? 0.0f : d + tr * XFERT;
    dist[t] = tt;                      // trip_times, in place

    a_dt  += de * tt;
    a_tr  += de * tr;
    a_dem += de;
    if (nopath) { a_un += de; if (de > 0.0f) a_nd += 1.0f; }
    if (tr == 0.0f)      a_b0 += de;
    else if (tr == 1.0f) a_b1 += de;
    else if (tr == 2.0f) a_b2 += de;
    if (tr > 2.0f)       a_bu += de;
  }

  float q[9] = {a_dt, a_tr, a_un, a_dem, a_nd, a_b0, a_b1, a_b2, a_bu};

  __shared__ float red[8][16];
  __shared__ float fin[16];
  const int wave = threadIdx.x >> 5;
  const int lane = threadIdx.x & 31;

  #pragma unroll
  for (int i = 0; i < 9; ++i) {
    float s = wmma_allreduce_f32(q[i]);   // all lanes active: EXEC all-1s
    if (lane == 0) red[wave][i] = s;
  }
  __syncthreads();

  if (threadIdx.x < 9) {
    float s = 0.0f;
    for (int w = 0; w < 8; ++w) s += red[w][threadIdx.x];
    fin[threadIdx.x] = s;
  }
  __syncthreads();

  if (threadIdx.x == 0) {
    out[0   + b] = fin[0];                       // total_demand_time
    out[128 + b] = fin[1];                       // total_transfers
    out[112 + b] = fin[2];                       // unserved_demand
    out[96  + b] = fin[3];                       // total_demand
    out[SCALARS + (size_t)BB * NN * NN + b] = fin[4];   // n_disconnected
    out[32 + b * 4 + 0] = fin[5];                // bucket 0
    out[32 + b * 4 + 1] = fin[6];                // bucket 1
    out[32 + b * 4 + 2] = fin[7];                // bucket 2
    out[32 + b * 4 + 3] = fin[8] + fin[2];       // >2 transfers + unserved
  }
}

// ---------------------------------------------------------------------------
extern "C" void kernel_launch(void* const* d_in, const int* in_sizes, int n_in,
                              void* d_out, int out_size, void* d_ws,
                              size_t ws_size, hipStream_t stream) {
  const float* drive  = (const float*)d_in[0];  // (B,N,N)
  const float* demand = (const float*)d_in[1];  // (B,N,N)
  const int*   routes = (const int*)d_in[2];    // (B,R,L)

  float* out = (float*)d_out;
  float* dist = out + SCALARS;                  // (B,N,N) dist -> trip_times
  unsigned* dist_bits = (unsigned*)dist;
  unsigned char* hops = (unsigned char*)d_ws;   // 1 MB of scratch

  init_kernel<<<(BB * NN * NN + 255) / 256, 256, 0, stream>>>(out, dist_bits);
  routes_kernel<<<BB * RR, 32, 0, stream>>>(drive, routes, dist_bits);
  route_time_kernel<<<BB, 32, 0, stream>>>(drive, routes, out + 16);
  fw_kernel<<<BB, 1024, 0, stream>>>(dist, hops);
  finalize_kernel<<<BB, 256, 0, stream>>>(demand, out, hops);
}